// DWT2D_14104672600640
// MI455X (gfx1250) — compile-verified
//
#include <hip/hip_runtime.h>

// DWT2D Haar (B=8, N=512, C=16) -> out (8,256,256,64), channel order [LL,LH,HL,HH].
//
// Roofline: ~0.13 GFLOP vs 256 MiB of traffic -> AI ~0.5 flop/B, i.e. purely
// HBM-bound with a ~11.5 us floor at 23.3 TB/s. The Haar analysis matrix has
// exactly 2 nonzeros per row, so y = A x A^T collapses to a 2x2 butterfly per
// output pixel; WMMA would inflate FLOPs ~250x and force 4B-granularity access
// (C=16 is innermost). The chip-rewarded path is data movement:
//   - gfx1250 async DMA global->LDS (global_load_async_to_lds_b128, ASYNCcnt)
//     with non-temporal hint (read-once input, in+out > 192 MiB L2),
//   - conflict-free ds_load_b128 (4 addresses x 4-lane broadcast per half-wave),
//   - perfectly contiguous non-temporal b128 stores.

typedef float f4 __attribute__((ext_vector_type(4)));

#define THREADS 256
#define TILE_F4 1024      // 2 input rows * 128 pixels * 4 float4 = 16 KiB
#define ITERS   4         // 1024 output float4 per block / 256 threads

__global__ __launch_bounds__(THREADS)
void dwt2d_haar_kernel(const f4* __restrict__ x, f4* __restrict__ out)
{
    __shared__ f4 tile[TILE_F4];   // [0..511]=row 2i segment, [512..1023]=row 2i+1

    const int      t   = threadIdx.x;
    const unsigned bid = blockIdx.x;
    const unsigned seg = bid & 3u;        // 4 segments of 64 output pixels per row
    const unsigned row = bid >> 2;        // row = b*256 + i
    const unsigned i   = row & 255u;
    const unsigned b   = row >> 8;
    const unsigned l0  = seg * 64u;

    // float4 indices into global memory
    const size_t g_row0 = ((size_t)(b * 512u + 2u * i) * 512u + 2u * l0) * 4u;
    const size_t g_row1 = g_row0 + 2048u;                  // next input row (512 px * 4 f4)
    const size_t g_out  = ((size_t)row * 256u + l0) * 16u; // 64 out pixels * 16 f4

    // Low 32 bits of the flat address of an LDS object are the LDS byte offset
    // (shared aperture lives entirely in addr[63:32] per CDNA5 aperture rules).
    const unsigned lds_base = (unsigned)(size_t)&tile[0];

    // ---- Stage 16 KiB input tile via async DMA to LDS (ASYNCcnt path) ----
    // k selects the input row at compile time (loop is fully unrolled), so the
    // address select folds and every async issue runs with full EXEC.
#if defined(__HIP_DEVICE_COMPILE__)
#pragma unroll
    for (int k = 0; k < 4; ++k) {
        const size_t  gbase = (k < 2) ? g_row0 : (g_row1 - 512u);
        const unsigned f    = (unsigned)(k * THREADS) + (unsigned)t;  // 0..1023
        const f4*     gptr  = x + (gbase + f);                        // 512B/wave, coalesced
        const unsigned loff = lds_base + f * 16u;
        asm volatile("global_load_async_to_lds_b128 %0, %1, off th:TH_LOAD_NT"
                     :: "v"(loff), "v"(gptr) : "memory");
    }
    asm volatile("s_wait_asynccnt 0" ::: "memory");
#endif
    __syncthreads();

    // ---- Butterfly + contiguous non-temporal stores ----
    const float s = 0.70710678118654752440f;  // 1/sqrt(2), matches reference filters
#pragma unroll
    for (int it = 0; it < ITERS; ++it) {
        int o   = it * THREADS + t;  // local output float4 index, 0..1023 (contiguous)
        int p   = o >> 4;            // local output pixel 0..63
        int sub = o & 15;
        int qd  = sub >> 2;          // 0=LL 1=LH 2=HL 3=HH
        int r   = sub & 3;           // channel float4 within pixel

        // 4 distinct addresses x 4-lane broadcast per half-wave: conflict-free
        f4 a  = tile[8 * p + r];            // x[2i,   2l,   4r..]
        f4 bb = tile[8 * p + 4 + r];        // x[2i,   2l+1, 4r..]
        f4 c  = tile[512 + 8 * p + r];      // x[2i+1, 2l,   4r..]
        f4 d  = tile[512 + 8 * p + 4 + r];  // x[2i+1, 2l+1, 4r..]

        float cs = (qd & 2) ? -1.0f : 1.0f; // column low/high pass
        float rs = (qd & 1) ? -1.0f : 1.0f; // row    low/high pass

        f4 u   = (a + cs * bb) * s;         // column transform, row 2i
        f4 v   = (c + cs * d)  * s;         // column transform, row 2i+1
        f4 res = (u + rs * v)  * s;         // row transform (two-stage, mirrors reference)

        __builtin_nontemporal_store(res, &out[g_out + (size_t)o]); // 512B/wave contiguous
    }
}

extern "C" void kernel_launch(void* const* d_in, const int* in_sizes, int n_in,
                              void* d_out, int out_size, void* d_ws, size_t ws_size,
                              hipStream_t stream) {
    (void)in_sizes; (void)n_in; (void)d_ws; (void)ws_size; (void)out_size;
    const f4* x  = (const f4*)d_in[0];      // (8,512,512,16) f32
    // d_in[1] = A is the deterministic Haar analysis matrix; its 2-nonzero/row
    // structure is folded into the butterfly, so it is not read.
    f4* out = (f4*)d_out;                   // (8,256,256,64) f32
    const int blocks = 8 * 256 * 4;         // B * m * (m/64)
    dwt2d_haar_kernel<<<blocks, THREADS, 0, stream>>>(x, out);
}